// PairwiseAUCLoss_51110110822862
// MI455X (gfx1250) — compile-verified
//
#include <hip/hip_runtime.h>
#include <hip/hip_bf16.h>
#include <math.h>

// ---------------------------------------------------------------------------
// PairwiseAUCLoss on MI455X (gfx1250, wave32)
//
// Phase 1: order-preserving first-k selection, one workgroup per class.
//          Ballot + block prefix ranks; early exit once 32 pos / 64 neg found
//          (typical traffic ~100KB instead of streaming 235MB).
// Phase 2: 14 waves in one block; per class build the 32x64 difference matrix
//          diff = pos*1^T - 1*neg^T with V_WMMA_F32_16X16X4_F32 (rank-2 GEMM,
//          exact f32), then softplus + wave reduction + class mean.
// ---------------------------------------------------------------------------

#define NCLS    14
#define MAXPOS  32
#define MAXNEG  64
#define SCAN_BLK 1024

typedef __attribute__((ext_vector_type(2))) float v2f;
typedef __attribute__((ext_vector_type(8))) float v8f;

__device__ __forceinline__ float softplus_stable(float x) {
  // softplus(x) = max(x,0) + log1p(exp(-|x|)); exact 0 for x = -1e30 sentinels
  return fmaxf(x, 0.0f) + log1pf(expf(-fabsf(x)));
}

// ---------------------------------------------------------------------------
// Kernel 1: per-class deterministic first-k selection (batch order preserved).
// grid = 14 blocks (one class each), block = 1024 threads = 32 waves.
// ---------------------------------------------------------------------------
__global__ __launch_bounds__(SCAN_BLK) void firstk_select_kernel(
    const float* __restrict__ logits, const int* __restrict__ targets,
    long long B, float* __restrict__ posbuf, float* __restrict__ negbuf,
    int* __restrict__ meta)
{
  const int c    = blockIdx.x;
  const int tid  = threadIdx.x;
  const int lane = tid & 31;
  const int wv   = tid >> 5;          // wave index 0..31

  __shared__ int s_wp[32];            // per-wave positive counts
  __shared__ int s_wn[32];            // per-wave negative counts
  __shared__ int s_pbase, s_nbase, s_done;

  // Sentinel init: unfilled pos slots -> +1e30 (diff=+big -> softplus(-big)=0)
  //                unfilled neg slots -> -1e30 (diff=+big -> softplus(-big)=0)
  if (tid < MAXPOS) posbuf[c * MAXPOS + tid] =  1.0e30f;
  if (tid < MAXNEG) negbuf[c * MAXNEG + tid] = -1.0e30f;
  if (tid == 0) { s_pbase = 0; s_nbase = 0; s_done = 0; }
  __syncthreads();

  for (long long base = 0; base < B; base += SCAN_BLK) {
    const long long r = base + tid;
    int t = -1;
    if (r < B) {
      t = targets[r * NCLS + c];
      const long long rn = r + SCAN_BLK;
      if (rn < B) __builtin_prefetch(&targets[rn * NCLS + c], 0, 1);
    }
    const bool isPos = (t == 1);
    const bool isNeg = (t == 0);

    // wave32 ballot ranks (order-preserving within wave)
    const unsigned pm = (unsigned)__ballot(isPos);
    const unsigned nm = (unsigned)__ballot(isNeg);
    const unsigned lowmask = (1u << lane) - 1u;
    const int rkP = __popc(pm & lowmask);
    const int rkN = __popc(nm & lowmask);
    if (lane == 0) { s_wp[wv] = __popc(pm); s_wn[wv] = __popc(nm); }
    __syncthreads();

    // exclusive prefix over the 32 wave counts (redundant per thread, tiny)
    int offP = 0, offN = 0;
    for (int w = 0; w < wv; ++w) { offP += s_wp[w]; offN += s_wn[w]; }
    const int pbase = s_pbase, nbase = s_nbase;
    const int gP = pbase + offP + rkP;
    const int gN = nbase + offN + rkN;
    if (isPos && gP < MAXPOS) posbuf[c * MAXPOS + gP] = logits[r * NCLS + c];
    if (isNeg && gN < MAXNEG) negbuf[c * MAXNEG + gN] = logits[r * NCLS + c];
    __syncthreads();

    if (tid == 0) {
      int tp = 0, tn = 0;
      for (int w = 0; w < 32; ++w) { tp += s_wp[w]; tn += s_wn[w]; }
      int np = pbase + tp; if (np > MAXPOS) np = MAXPOS;  // saturate: take=min(n,k)
      int nn = nbase + tn; if (nn > MAXNEG) nn = MAXNEG;
      s_pbase = np; s_nbase = nn;
      s_done = (np >= MAXPOS && nn >= MAXNEG) ? 1 : 0;
    }
    __syncthreads();
    if (s_done) break;                 // uniform early exit (typical: 1 iter)
  }

  if (tid == 0) {
    meta[c * 2 + 0] = s_pbase;         // take_pos = min(n_pos, 32)
    meta[c * 2 + 1] = s_nbase;         // take_neg = min(n_neg, 64)
  }
}

// ---------------------------------------------------------------------------
// Kernel 2: pairwise softplus loss via WMMA rank-2 matrix build.
// One block, 14 waves (one class per wave), wave32.
// diff tile (16x16) = A(16x4) x B(4x16):
//   A[i][0] = pos[i], A[i][1] = 1,      A[i][2..3] = 0
//   B[0][j] = 1,      B[1][j] = -neg[j], B[2..3][j] = 0
// => D[i][j] = pos[i] - neg[j]   (MARGIN == 0), exact f32.
// ---------------------------------------------------------------------------
__global__ __launch_bounds__(NCLS * 32) void pairwise_auc_kernel(
    const float* __restrict__ posbuf, const float* __restrict__ negbuf,
    const int* __restrict__ meta, float* __restrict__ out)
{
  const int c    = threadIdx.x >> 5;   // class == wave id
  const int lane = threadIdx.x & 31;
  const int l15  = lane & 15;
  const bool lo  = lane < 16;

  __shared__ float s_loss[NCLS];
  __shared__ int   s_act[NCLS];

  float acc = 0.0f;
#pragma unroll
  for (int mt = 0; mt < 2; ++mt) {     // 32 pos rows = 2 tiles of 16
    // A layout (16x4 f32): VGPR0 = {K=0 | K=2}, VGPR1 = {K=1 | K=3}
    const float p = posbuf[c * MAXPOS + mt * 16 + l15];
    v2f a;
    a.x = lo ? p    : 0.0f;            // K=0 col = pos ; K=2 col = 0
    a.y = lo ? 1.0f : 0.0f;            // K=1 col = 1   ; K=3 col = 0
#pragma unroll
    for (int nt = 0; nt < 4; ++nt) {   // 64 neg cols = 4 tiles of 16
      const float n = negbuf[c * MAXNEG + nt * 16 + l15];
      v2f b;
      b.x = lo ? 1.0f : 0.0f;          // B row 0 = ones ; row 2 = 0
      b.y = lo ? (-n) : 0.0f;          // B row 1 = -neg ; row 3 = 0
      v8f cz = {};
      // 8 args: (neg_a, A, neg_b, B, c_mod, C, reuse_a, reuse_b)
      v8f d = __builtin_amdgcn_wmma_f32_16x16x4_f32(
          false, a, false, b, (short)0, cz, false, false);
#pragma unroll
      for (int k = 0; k < 8; ++k) acc += softplus_stable(-d[k]);
    }
  }

  // wave32 reduction of the 16x16 tile partial sums
#pragma unroll
  for (int off = 16; off > 0; off >>= 1) acc += __shfl_xor(acc, off, 32);

  if (lane == 0) {
    const int np = meta[c * 2 + 0];
    const int nn = meta[c * 2 + 1];
    const bool active = (np > 0) && (nn > 0);
    const float pairs = (float)(np * nn);
    s_loss[c] = active ? (acc / fmaxf(pairs, 1.0f)) : 0.0f;
    s_act[c]  = active ? 1 : 0;
  }
  __syncthreads();
  if (threadIdx.x == 0) {
    float s = 0.0f; int cnt = 0;
    for (int i = 0; i < NCLS; ++i) { s += s_loss[i]; cnt += s_act[i]; }
    out[0] = (cnt > 0) ? (s / (float)cnt) : 0.0f;
  }
}

// ---------------------------------------------------------------------------
extern "C" void kernel_launch(void* const* d_in, const int* in_sizes, int n_in,
                              void* d_out, int out_size, void* d_ws, size_t ws_size,
                              hipStream_t stream) {
  const float* logits  = (const float*)d_in[0];
  const int*   targets = (const int*)d_in[1];
  const long long B = (long long)in_sizes[0] / NCLS;

  char* ws = (char*)d_ws;
  float* posbuf = (float*)ws;                                            // 14*32 f32
  float* negbuf = (float*)(ws + NCLS * MAXPOS * sizeof(float));          // 14*64 f32
  int*   meta   = (int*)(ws + NCLS * (MAXPOS + MAXNEG) * sizeof(float)); // 14*2 i32

  firstk_select_kernel<<<dim3(NCLS), dim3(SCAN_BLK), 0, stream>>>(
      logits, targets, B, posbuf, negbuf, meta);
  pairwise_auc_kernel<<<dim3(1), dim3(NCLS * 32), 0, stream>>>(
      posbuf, negbuf, meta, (float*)d_out);
}